// LSTMDecoder_85590108275014
// MI455X (gfx1250) — compile-verified
//
#include <hip/hip_runtime.h>
#include <hip/hip_bf16.h>

typedef __attribute__((ext_vector_type(2))) float v2f;
typedef __attribute__((ext_vector_type(8))) float v8f;

// D = A(16x4) * B(4x16) + C, fp32 exact. ISA: V_WMMA_F32_16X16X4_F32.
__device__ __forceinline__ v8f wmma4(v2f a, v2f b, v8f c) {
  return __builtin_amdgcn_wmma_f32_16x16x4_f32(false, a, false, b, (short)0, c,
                                               false, false);
}

// ---------------------------------------------------------------------------
// pre[b][r][h] = sum_f feats[b][r][f] * Wf[h][f] + b1[h]
// feats[b][r][f] = features[b][f][r]  (features: B x 2048 x 49)
// Wf = attn_W1[:, :2048] (row-major H x 2560) == N x K layout for B-fragment.
// One wave computes a 16(M) x 64(N) strip. 2048 waves total.
// ---------------------------------------------------------------------------
__global__ __launch_bounds__(256) void wmma_pre_kernel(
    const float* __restrict__ features, const float* __restrict__ attn_W1,
    const float* __restrict__ attn_b1, float* __restrict__ pre) {
  int wid = threadIdx.x >> 5, lane = threadIdx.x & 31;
  int lin = blockIdx.x * 8 + wid;  // 0..2047
  int b = lin >> 5;
  int rem = lin & 31;
  int mt = rem >> 3;    // 4 row tiles cover r=0..63 (49 valid)
  int strip = rem & 7;  // 8 strips of 64 cols cover H=512
  int n0 = strip * 64, m0 = mt * 16;
  int half = lane >> 4, mr = lane & 15;
  int r = m0 + mr;
  if (r > 48) r = 48;  // clamp A reads; tail rows discarded at store
  const float* fb = features + (size_t)b * 2048 * 49;

  v8f acc[4] = {};
  for (int k = 0; k < 2048; k += 4) {
    int kk = k + 2 * half;
    v2f a;
    a.x = fb[(size_t)kk * 49 + r];
    a.y = fb[(size_t)(kk + 1) * 49 + r];
#pragma unroll
    for (int j = 0; j < 4; ++j) {
      v2f bf = *(const v2f*)(attn_W1 + (size_t)(n0 + j * 16 + mr) * 2560 + kk);
      acc[j] = wmma4(a, bf, acc[j]);
    }
  }
#pragma unroll
  for (int j = 0; j < 4; ++j) {
    int n = n0 + j * 16 + mr;
    float bias = attn_b1[n];
#pragma unroll
    for (int v = 0; v < 8; ++v) {
      int row = m0 + v + 8 * half;
      if (row < 49) pre[((size_t)b * 49 + row) * 512 + n] = acc[j][v] + bias;
    }
  }
}

// ---------------------------------------------------------------------------
// Generic dual-source GEMM:  C[M x N] = A1*B1t^T (+ A2*B2t^T) + bias1 + bias2
// A: row-major M x K.  Bt: row-major N x K (i.e. torch weight layout W[n][k]).
// Wave computes 16(M) x 64(N); A fragment reused across 4 accumulators.
// Whole-wave-uniform early exit keeps EXEC all-ones for WMMA.
// ---------------------------------------------------------------------------
__global__ __launch_bounds__(256) void wmma_gemm_nt(
    const float* __restrict__ A1, int lda1, const float* __restrict__ B1t,
    int ldb1, int K1len, const float* __restrict__ A2, int lda2,
    const float* __restrict__ B2t, int ldb2, int K2len,
    const float* __restrict__ bias1, const float* __restrict__ bias2,
    float* __restrict__ C, long ldc, int mtiles, int N) {
  int nstrips = (N + 63) >> 6;
  int wid = threadIdx.x >> 5, lane = threadIdx.x & 31;
  int lin = blockIdx.x * 8 + wid;
  if (lin >= mtiles * nstrips) return;  // wave-uniform
  int strip = lin % nstrips;
  int mt = lin / nstrips;
  int n0 = strip * 64, m0 = mt * 16;
  int half = lane >> 4, mr = lane & 15;  // mr: A row within tile / B col
  int jcount = (N - n0) >> 4;
  if (jcount > 4) jcount = 4;

  v8f acc[4] = {};
  const float* arow1 = A1 + (size_t)(m0 + mr) * lda1;
  for (int k = 0; k < K1len; k += 4) {
    v2f a = *(const v2f*)(arow1 + k + 2 * half);
#pragma unroll
    for (int j = 0; j < 4; ++j) {
      if (j < jcount) {
        v2f b =
            *(const v2f*)(B1t + (size_t)(n0 + j * 16 + mr) * ldb1 + k + 2 * half);
        acc[j] = wmma4(a, b, acc[j]);
      }
    }
  }
  if (A2) {
    const float* arow2 = A2 + (size_t)(m0 + mr) * lda2;
    for (int k = 0; k < K2len; k += 4) {
      v2f a = *(const v2f*)(arow2 + k + 2 * half);
#pragma unroll
      for (int j = 0; j < 4; ++j) {
        if (j < jcount) {
          v2f b = *(const v2f*)(B2t + (size_t)(n0 + j * 16 + mr) * ldb2 + k +
                                2 * half);
          acc[j] = wmma4(a, b, acc[j]);
        }
      }
    }
  }
  for (int j = 0; j < jcount; ++j) {
    int n = n0 + j * 16 + mr;
    float bsum = 0.f;
    if (bias1) bsum += bias1[n];
    if (bias2) bsum += bias2[n];
#pragma unroll
    for (int v = 0; v < 8; ++v) {
      size_t m = (size_t)(m0 + v + 8 * half);
      C[m * (size_t)ldc + n] = acc[j][v] + bsum;
    }
  }
}

// ---------------------------------------------------------------------------
// Per-batch attention: scores -> softmax -> context; also builds xin and
// writes attention weights. One block per batch element.
// ---------------------------------------------------------------------------
__global__ __launch_bounds__(256) void attn_ctx_kernel(
    const float* __restrict__ pre, const float* __restrict__ hWh,
    const float* __restrict__ features, const float* __restrict__ attn_W2,
    const float* __restrict__ attn_b2, const float* __restrict__ embedding,
    const int* __restrict__ captions, int t, float* __restrict__ xin,
    float* __restrict__ att_out) {
  __shared__ float s_scores[64];
  __shared__ float s_aw[64];
  int b = blockIdx.x;
  int tid = threadIdx.x;
  int wid = tid >> 5, lane = tid & 31;
  const float* preb = pre + (size_t)b * 49 * 512;
  const float* hwb = hWh + (size_t)b * 512;

  // scores[r] = sum_h tanh(pre[b][r][h] + (h*Wh^T)[b][h]) * w2[h] + b2
  for (int r = wid; r < 49; r += 8) {
    float p = 0.f;
    const float* prer = preb + (size_t)r * 512;
    for (int h = lane; h < 512; h += 32)
      p += tanhf(prer[h] + hwb[h]) * attn_W2[h];
#pragma unroll
    for (int off = 16; off > 0; off >>= 1) p += __shfl_xor(p, off);
    if (lane == 0) s_scores[r] = p + attn_b2[0];
  }
  __syncthreads();
  if (tid == 0) {  // softmax over 49 — trivial, serial
    float mx = s_scores[0];
    for (int r = 1; r < 49; ++r) mx = fmaxf(mx, s_scores[r]);
    float sum = 0.f;
    for (int r = 0; r < 49; ++r) {
      float e = expf(s_scores[r] - mx);
      s_aw[r] = e;
      sum += e;
    }
    float inv = 1.f / sum;
    for (int r = 0; r < 49; ++r) s_aw[r] *= inv;
  }
  __syncthreads();
  if (tid < 49) att_out[((size_t)b * 32 + t) * 49 + tid] = s_aw[tid];

  // xin[:, 0:512] = embedding[captions[b][t]]
  int cap = captions[b * 32 + t];
  float* xb = xin + (size_t)b * 2560;
  for (int e = tid; e < 512; e += 256) xb[e] = embedding[(size_t)cap * 512 + e];

  // xin[:, 512:] = ctx[f] = sum_r aw[r] * features[b][f][r] (r contiguous)
  const float* fb = features + (size_t)b * 2048 * 49;
  for (int f = tid; f < 2048; f += 256) {
    const float* fr = fb + (size_t)f * 49;
    float acc = 0.f;
    for (int r = 0; r < 49; ++r) acc += s_aw[r] * fr[r];
    xb[512 + f] = acc;
  }
}

// ---------------------------------------------------------------------------
// LSTM cell elementwise update. 64x512 threads.
// ---------------------------------------------------------------------------
__global__ void lstm_cell_kernel(const float* __restrict__ gates,
                                 float* __restrict__ h, float* __restrict__ c) {
  int idx = blockIdx.x * blockDim.x + threadIdx.x;
  if (idx >= 64 * 512) return;
  int b = idx >> 9, hh = idx & 511;
  const float* g = gates + (size_t)b * 2048;
  float ig = g[hh], fg = g[512 + hh], gg = g[1024 + hh], og = g[1536 + hh];
  float si = 1.f / (1.f + expf(-ig));
  float sf = 1.f / (1.f + expf(-fg));
  float so = 1.f / (1.f + expf(-og));
  float cn = sf * c[idx] + si * tanhf(gg);
  float hn = so * tanhf(cn);
  c[idx] = cn;
  h[idx] = hn;
}

__global__ void zero_kernel(float* __restrict__ p, int n) {
  int i = blockIdx.x * blockDim.x + threadIdx.x;
  if (i < n) p[i] = 0.f;
}

// ---------------------------------------------------------------------------
extern "C" void kernel_launch(void* const* d_in, const int* in_sizes, int n_in,
                              void* d_out, int out_size, void* d_ws,
                              size_t ws_size, hipStream_t stream) {
  const float* features = (const float*)d_in[0];    // (64,2048,7,7)
  const int* captions = (const int*)d_in[1];        // (64,32)
  const float* embedding = (const float*)d_in[2];   // (10000,512)
  const float* attn_W1 = (const float*)d_in[3];     // (512,2560)
  const float* attn_b1 = (const float*)d_in[4];     // (512)
  const float* attn_W2 = (const float*)d_in[5];     // (1,512)
  const float* attn_b2 = (const float*)d_in[6];     // (1)
  const float* W_ih = (const float*)d_in[7];        // (2048,2560)
  const float* W_hh = (const float*)d_in[8];        // (2048,512)
  const float* b_ih = (const float*)d_in[9];        // (2048)
  const float* b_hh = (const float*)d_in[10];       // (2048)
  const float* fc_W = (const float*)d_in[11];       // (10000,512)
  const float* fc_b = (const float*)d_in[12];       // (10000)

  float* out = (float*)d_out;                       // (64,32,10000)
  float* att_out = out + (size_t)64 * 32 * 10000;   // (64,32,49)

  float* ws = (float*)d_ws;
  float* pre = ws;                        // 64*49*512
  float* hWh = pre + (size_t)64 * 49 * 512;  // 64*512
  float* xin = hWh + 64 * 512;            // 64*2560
  float* gates = xin + 64 * 2560;         // 64*2048
  float* hbuf = gates + 64 * 2048;        // 64*512
  float* cbuf = hbuf + 64 * 512;          // 64*512 (contiguous after hbuf)

  // h0 = c0 = 0 (hbuf,cbuf contiguous: 65536 floats)
  zero_kernel<<<(65536 + 255) / 256, 256, 0, stream>>>(hbuf, 65536);

  // pre = feats @ Wf^T + b1  (once; 2048 waves)
  wmma_pre_kernel<<<256, 256, 0, stream>>>(features, attn_W1, attn_b1, pre);

  for (int t = 0; t < 32; ++t) {
    // hWh = h @ Wh^T   (M=64, K=512, N=512) -> 32 waves
    wmma_gemm_nt<<<4, 256, 0, stream>>>(hbuf, 512, attn_W1 + 2048, 2560, 512,
                                        nullptr, 0, nullptr, 0, 0, nullptr,
                                        nullptr, hWh, 512, 4, 512);
    // attention + softmax + context + xin assembly
    attn_ctx_kernel<<<64, 256, 0, stream>>>(pre, hWh, features, attn_W2,
                                            attn_b2, embedding, captions, t,
                                            xin, att_out);
    // gates = xin @ W_ih^T + h @ W_hh^T + b_ih + b_hh  (M=64, N=2048) -> 128 waves
    wmma_gemm_nt<<<16, 256, 0, stream>>>(xin, 2560, W_ih, 2560, 2560, hbuf, 512,
                                         W_hh, 512, 512, b_ih, b_hh, gates,
                                         2048, 4, 2048);
    // LSTM cell -> h_new, c_new
    lstm_cell_kernel<<<128, 256, 0, stream>>>(gates, hbuf, cbuf);
    // out[:, t, :] = h_new @ fc_W^T + fc_b  (M=64, N=10000) -> 628 waves
    wmma_gemm_nt<<<79, 256, 0, stream>>>(hbuf, 512, fc_W, 512, 512, nullptr, 0,
                                         nullptr, 0, 0, fc_b, nullptr,
                                         out + (size_t)t * 10000,
                                         (long)32 * 10000, 4, 10000);
  }
}